// TransformationLoss_5669356835830
// MI455X (gfx1250) — compile-verified
//
#include <hip/hip_runtime.h>
#include <math.h>

#define BS      32
#define NPTS    8192
#define NCHUNK  32          // points per block
#define RE_THRE 15.0f
#define TE_THRE 30.0f

typedef __attribute__((ext_vector_type(2))) float v2f;
typedef __attribute__((ext_vector_type(8))) float v8f;

// Raw hardware sqrt (v_sqrt_f32, ~1ulp) — avoids the correctly-rounded
// library expansion (~20 VALU ops each). TRANS op, co-executes with VALU.
__device__ __forceinline__ float fast_sqrtf(float x) {
    return __builtin_amdgcn_sqrtf(x);
}

// ws layout (floats):
// [0,32)   lsum per b   (sum of sq over b2,n)
// [32,64)  rsum per b   (sum of sqrt(sq) over b2,n)
// [64,96)  re per b
// [96,128) te per b
// [128,160) has_inlier flag per b

__global__ void zero_ws_kernel(float* ws) {
    ws[threadIdx.x] = 0.0f;   // 160 threads
}

// One block per batch b: re, te, has_inlier
__global__ void stats_kernel(const float* __restrict__ trans,
                             const float* __restrict__ gt_trans,
                             const float* __restrict__ probs,
                             float* __restrict__ ws) {
    const int b = blockIdx.x;
    __shared__ int any_flag;
    if (threadIdx.x == 0) any_flag = 0;
    __syncthreads();

    int local = 0;
    for (int i = threadIdx.x; i < NPTS; i += blockDim.x)
        local |= (probs[b * NPTS + i] > 0.0f) ? 1 : 0;
    if (local) atomicOr(&any_flag, 1);
    __syncthreads();

    if (threadIdx.x == 0) {
        const float* T = trans + b * 16;
        const float* G = gt_trans + b * 16;
        float tr = 0.0f, te2 = 0.0f;
        #pragma unroll
        for (int i = 0; i < 3; ++i) {
            #pragma unroll
            for (int j = 0; j < 3; ++j) tr += T[i * 4 + j] * G[i * 4 + j];
            float d = T[i * 4 + 3] - G[i * 4 + 3];
            te2 += d * d;
        }
        float c = (tr - 1.0f) * 0.5f;
        c = fminf(1.0f, fmaxf(-1.0f, c));
        ws[64 + b]  = acosf(c) * 57.29577951308232f;   // 180/pi
        ws[96 + b]  = sqrtf(te2) * 100.0f;
        ws[128 + b] = any_flag ? 1.0f : 0.0f;
    }
}

// Main fused cross-distance kernel.
// grid = NPTS/NCHUNK blocks of 256 threads (8 waves).
__global__ __launch_bounds__(256) void cross_kernel(const float* __restrict__ trans,
                                                    const float* __restrict__ src,
                                                    const float* __restrict__ tgt,
                                                    float* __restrict__ ws) {
    // sA[n][b] = (Wx, Wy, Wz, |W|^2) ; sB[n][b2] = (-2Tx, -2Ty, -2Tz, |T|^2)
    __shared__ float4 sA[NCHUNK][BS];
    __shared__ float4 sB[NCHUNK][BS];
    __shared__ float accumBlk[64];           // [0,32)=loss  [32,64)=rmse

    const int tid = threadIdx.x;
    const int n0  = blockIdx.x * NCHUNK;

    if (tid < 64) accumBlk[tid] = 0.0f;

    // ---- Phase 1: transform points, build WMMA-ready operand data in LDS ----
    {
        const int b = tid >> 3;                // fixed batch per thread
        const float* Tb = trans + b * 16;
        const float r00 = Tb[0], r01 = Tb[1], r02 = Tb[2],  t0 = Tb[3];
        const float r10 = Tb[4], r11 = Tb[5], r12 = Tb[6],  t1 = Tb[7];
        const float r20 = Tb[8], r21 = Tb[9], r22 = Tb[10], t2 = Tb[11];
        #pragma unroll
        for (int i = 0; i < NCHUNK / 8; ++i) {
            const int nl = (tid & 7) + 8 * i;
            const int n  = n0 + nl;
            const float* s = src + ((size_t)b * NPTS + n) * 3;
            const float* g = tgt + ((size_t)b * NPTS + n) * 3;
            // Pull next chunk toward this WGP while we compute this one.
            if (n + NCHUNK < NPTS) {
                __builtin_prefetch(s + NCHUNK * 3, 0, 3);
                __builtin_prefetch(g + NCHUNK * 3, 0, 3);
            }
            const float x = s[0], y = s[1], z = s[2];
            const float wx = r00 * x + r01 * y + r02 * z + t0;
            const float wy = r10 * x + r11 * y + r12 * z + t1;
            const float wz = r20 * x + r21 * y + r22 * z + t2;
            sA[nl][b] = make_float4(wx, wy, wz, wx * wx + wy * wy + wz * wz);
            const float gx = g[0], gy = g[1], gz = g[2];
            sB[nl][b] = make_float4(-2.0f * gx, -2.0f * gy, -2.0f * gz,
                                    gx * gx + gy * gy + gz * gz);
        }
    }
    __syncthreads();

    // ---- Phase 2: 32x32 pair matrix per point via 4x V_WMMA_F32_16X16X4_F32 ----
    const int wave = tid >> 5;
    const int lane = tid & 31;
    const int half = lane >> 4;        // 0/1: K-half for A/B, M-offset(+8) for C/D
    const int l16  = lane & 15;

    float lsum[4][8];
    float rsum[4][8];
    #pragma unroll
    for (int t = 0; t < 4; ++t)
        #pragma unroll
        for (int r = 0; r < 8; ++r) { lsum[t][r] = 0.0f; rsum[t][r] = 0.0f; }

    const int NPW = NCHUNK / 8;        // points per wave
    #pragma unroll
    for (int i = 0; i < NPW; ++i) {
        const int nl = wave * NPW + i;

        // A tile tb: row M = tb*16 + l16, K in {0,1}(half0) / {2,3}(half1)
        v2f a[2], bb[2];
        #pragma unroll
        for (int tb = 0; tb < 2; ++tb) {
            const float4 wv = sA[nl][tb * 16 + l16];
            a[tb].x = half ? wv.z : wv.x;     // K=0:Wx  | K=2:Wz
            a[tb].y = half ? 1.0f : wv.y;     // K=1:Wy  | K=3:1
        }
        // B tile cb: col N = cb*16 + l16
        #pragma unroll
        for (int cb = 0; cb < 2; ++cb) {
            const float4 tv = sB[nl][cb * 16 + l16];
            bb[cb].x = half ? tv.z : tv.x;    // K=0:-2Tx | K=2:-2Tz
            bb[cb].y = half ? tv.w : tv.y;    // K=1:-2Ty | K=3:|T|^2
        }

        #pragma unroll
        for (int tb = 0; tb < 2; ++tb) {
            #pragma unroll
            for (int cb = 0; cb < 2; ++cb) {
                v8f c = {};
                c = __builtin_amdgcn_wmma_f32_16x16x4_f32(
                        false, a[tb], false, bb[cb], (short)0, c, false, false);
                const int t4 = tb * 2 + cb;
                #pragma unroll
                for (int r = 0; r < 8; ++r) {
                    // D element: (b = tb*16 + r + 8*half, b2 = cb*16 + l16)
                    const float wsq = sA[nl][tb * 16 + r + 8 * half].w;
                    float sq = c[r] + wsq;
                    sq = fmaxf(sq, 0.0f);     // guard fp rounding before sqrt
                    lsum[t4][r] += sq;
                    rsum[t4][r] += fast_sqrtf(sq);
                }
            }
        }
    }

    // ---- reduce over b2 (16 lanes per half), accumulate into block LDS ----
    #pragma unroll
    for (int t4 = 0; t4 < 4; ++t4) {
        #pragma unroll
        for (int r = 0; r < 8; ++r) {
            float l = lsum[t4][r], s = rsum[t4][r];
            #pragma unroll
            for (int m = 1; m < 16; m <<= 1) {
                l += __shfl_xor(l, m, 32);
                s += __shfl_xor(s, m, 32);
            }
            if (l16 == 0) {
                const int b = (t4 >> 1) * 16 + r + 8 * half;
                atomicAdd(&accumBlk[b],      l);
                atomicAdd(&accumBlk[32 + b], s);
            }
        }
    }
    __syncthreads();
    if (tid < 64) atomicAdd(&ws[tid], accumBlk[tid]);
}

// Single wave: fold 32 batches into the 5 output scalars.
__global__ void finalize_kernel(const float* __restrict__ ws, float* __restrict__ out) {
    const int b = threadIdx.x;                       // 32 lanes
    const float inv = 1.0f / ((float)BS * (float)NPTS);
    const float lsum = ws[b] * inv;                  // loss_per[b]
    const float rsum = ws[32 + b] * inv;             // rmse_per[b]
    const float re   = ws[64 + b];
    const float te   = ws[96 + b];
    const float flag = ws[128 + b];

    float vals[5];
    vals[0] = (flag > 0.0f) ? lsum : 0.0f;
    vals[1] = ((te < TE_THRE) && (re < RE_THRE)) ? 1.0f : 0.0f;
    vals[2] = re;
    vals[3] = te;
    vals[4] = rsum;

    #pragma unroll
    for (int i = 0; i < 5; ++i) {
        float v = vals[i];
        #pragma unroll
        for (int m = 1; m < 32; m <<= 1) v += __shfl_xor(v, m, 32);
        vals[i] = v;
    }
    if (b == 0) {
        out[0] = vals[0] / (float)BS;                 // loss
        out[1] = vals[1] * 100.0f / (float)BS;        // recall
        out[2] = vals[2] / (float)BS;                 // re.mean
        out[3] = vals[3] / (float)BS;                 // te.mean
        out[4] = vals[4] / (float)BS;                 // rmse_per.mean
    }
}

extern "C" void kernel_launch(void* const* d_in, const int* in_sizes, int n_in,
                              void* d_out, int out_size, void* d_ws, size_t ws_size,
                              hipStream_t stream) {
    const float* trans    = (const float*)d_in[0];
    const float* gt_trans = (const float*)d_in[1];
    const float* src      = (const float*)d_in[2];
    const float* tgt      = (const float*)d_in[3];
    const float* probs    = (const float*)d_in[4];
    float* ws  = (float*)d_ws;
    float* out = (float*)d_out;

    zero_ws_kernel <<<1, 160, 0, stream>>>(ws);
    stats_kernel   <<<BS, 256, 0, stream>>>(trans, gt_trans, probs, ws);
    cross_kernel   <<<NPTS / NCHUNK, 256, 0, stream>>>(trans, src, tgt, ws);
    finalize_kernel<<<1, 32, 0, stream>>>(ws, out);
}